// RotT_77464030150792
// MI455X (gfx1250) — compile-verified
//
#include <hip/hip_runtime.h>

// Qutrit rotation R = expm(-0.5j * theta * S), S = |0><1| + |1><0| on a
// 15-qutrit statevector, target qudit 7.
//   out[a,0,d] = c*x[a,0,d] - i*s*x[a,1,d]
//   out[a,1,d] = -i*s*x[a,0,d] + c*x[a,1,d]
//   out[a,2,d] = x[a,2,d]                      (pure copy -> async LDS DMA)
// with c = cos(theta/2), s = sin(theta/2).
//
// Memory-bound: ~230 MB traffic -> ~10 us at 23.3 TB/s. WMMA deliberately
// NOT used (AI ~0.25 FLOP/B; a 3x3 gate in a 16x16 tile is >90% waste).
// CDNA5-specific path: t=2 plane moved with the async global<->LDS DMA
// engines (ASYNCcnt), bypassing VGPRs entirely.
//
// sin/cos are hoisted into a 1-thread prologue kernel so the main kernel's
// waves don't each execute the ~350-SALU scalarized argument-reduction blob.

#define P3_7 2187u          // 3^7  (t-stride in elements)
#define P3_8 6561u          // 3^8  (a-stride in elements)
#define NSITES 4782969u     // 3^14 (a,d) site pairs

__global__ void __launch_bounds__(32)
rot_coeff_kernel(const float* __restrict__ angle, float* __restrict__ cs)
{
    if (threadIdx.x == 0) {
        const float half = 0.5f * angle[0];
        cs[0] = cosf(half);
        cs[1] = sinf(half);
    }
}

__global__ void __launch_bounds__(256)
qutrit_rot_kernel(const float2* __restrict__ x,
                  const float*  __restrict__ cs,
                  float2*       __restrict__ out)
{
    const unsigned p = blockIdx.x * blockDim.x + threadIdx.x;
    if (p >= NSITES) return;

    // Uniform scalar loads of the precomputed gate coefficients.
    const float c = cs[0];
    const float s = cs[1];

    const unsigned a  = p / P3_7;
    const unsigned d  = p - a * P3_7;
    const unsigned i0 = a * P3_8 + d;      // t = 0
    const unsigned i1 = i0 + P3_7;         // t = 1
    const unsigned i2 = i1 + P3_7;         // t = 2

    // ---- 2x2 complex rotation on levels {0,1} (coalesced b64 + VALU) ----
    const float2 x0 = x[i0];
    const float2 x1 = x[i1];
    float2 o0, o1;
    o0.x = c * x0.x + s * x1.y;   // Re(c*x0 - i*s*x1)
    o0.y = c * x0.y - s * x1.x;   // Im
    o1.x = c * x1.x + s * x0.y;   // Re(-i*s*x0 + c*x1)
    o1.y = c * x1.y - s * x0.x;   // Im
    out[i0] = o0;
    out[i1] = o1;

    // ---- t = 2 plane: pass-through copy via async global<->LDS DMA ----
    // Each lane stages its 8-byte element in its own LDS slot (tid*8).
    // Dynamic LDS (2048 B/block) starts at offset 0.
    const unsigned lds_off = threadIdx.x * 8u;
    const unsigned gbyte   = i2 * 8u;      // < 2^31, fits 32-bit GVS offset

    asm volatile("global_load_async_to_lds_b64 %0, %1, %2"
                 :: "v"(lds_off), "v"(gbyte), "s"(x)
                 : "memory");
    // Wait for this wave's async load to land in LDS before the async store
    // reads it back (both tracked by ASYNCcnt).
    __builtin_amdgcn_s_wait_asynccnt(0);
    asm volatile("global_store_async_from_lds_b64 %0, %1, %2"
                 :: "v"(gbyte), "v"(lds_off), "s"(out)
                 : "memory");
    // s_endpgm performs an implicit wait-idle, so the trailing async store
    // completes before the wave retires.
}

extern "C" void kernel_launch(void* const* d_in, const int* in_sizes, int n_in,
                              void* d_out, int out_size, void* d_ws, size_t ws_size,
                              hipStream_t stream) {
    (void)in_sizes; (void)n_in; (void)ws_size; (void)out_size;

    const float2* x     = (const float2*)d_in[0];   // complex64 statevector
    const float*  angle = (const float*)d_in[1];    // 1-element f32
    // d_in[2] (S) unused: expm has an exact closed form for this generator.
    float2* out = (float2*)d_out;
    float*  cs  = (float*)d_ws;                     // {cos, sin} scratch

    rot_coeff_kernel<<<1, 32, 0, stream>>>(angle, cs);

    const unsigned threads = 256;
    const unsigned blocks  = (NSITES + threads - 1) / threads;   // 18684
    // 2048 B dynamic LDS per block = one 8-byte staging slot per thread.
    qutrit_rot_kernel<<<blocks, threads, threads * 8, stream>>>(x, cs, out);
}